// LiveniumJoint_41016937677059
// MI455X (gfx1250) — compile-verified
//
#include <hip/hip_runtime.h>
#include <cstdint>

constexpr int   D         = 768;
constexpr int   NA        = 3;
constexpr int   CH        = 6;      // 6 float4 chunks/lane * 32 lanes * 4 = 768
constexpr int   STEPS     = 6;
constexpr float BETA      = 20.0f;
constexpr float ALPHA     = 0.05f;
constexpr float MAX_NORM  = 10.0f;
constexpr float EPS_NORM  = 1e-12f;
constexpr float EPS_CLIP  = 1e-8f;

typedef int   v4i __attribute__((ext_vector_type(4)));
typedef float v4f __attribute__((ext_vector_type(4)));
typedef __attribute__((address_space(1))) v4i* gas_v4i_ptr;
typedef __attribute__((address_space(3))) v4i* lds_v4i_ptr;

__device__ __forceinline__ float dot4(v4f a, v4f b) {
  return a.x * b.x + a.y * b.y + a.z * b.z + a.w * b.w;
}

// wave32 all-lanes butterfly reduction (result broadcast to every lane)
__device__ __forceinline__ float wave_sum(float v) {
#pragma unroll
  for (int m = 16; m >= 1; m >>= 1) v += __shfl_xor(v, m, 32);
  return v;
}

// ---- CDNA5 async global->LDS copy (ASYNCcnt path) -------------------------
__device__ __forceinline__ void async_copy16(const float* gsrc, float* lds_dst) {
#if defined(__has_builtin) && __has_builtin(__builtin_amdgcn_global_load_async_to_lds_b128)
  __builtin_amdgcn_global_load_async_to_lds_b128(
      (gas_v4i_ptr)(void*)gsrc,
      (lds_v4i_ptr)(void*)lds_dst, 0, 0);
#else
  unsigned loff = (unsigned)(unsigned long long)
      (__attribute__((address_space(3))) void*)lds_dst;
  unsigned long long ga = (unsigned long long)(uintptr_t)gsrc;
  asm volatile("global_load_async_to_lds_b128 %0, %1, off"
               :: "v"(loff), "v"(ga) : "memory");
#endif
}

__device__ __forceinline__ void wait_async_all() {
#if defined(__has_builtin) && __has_builtin(__builtin_amdgcn_s_wait_asynccnt)
  __builtin_amdgcn_s_wait_asynccnt(0);
#else
  asm volatile("s_wait_asynccnt 0x0" ::: "memory");
#endif
}
// ---------------------------------------------------------------------------

__global__ __launch_bounds__(256) void livenium_kernel(
    const float* __restrict__ h0, const float* __restrict__ anchors,
    float* __restrict__ outh, float* __restrict__ outa, int nrows) {
  __shared__ __align__(16) float sA[NA * D];   // 9216 B

  const int tid = threadIdx.x;

  // Stage anchors (3x768 f32) into LDS via async-to-LDS, shared by all waves.
  for (int idx = tid; idx < (NA * D) / 4; idx += 256)
    async_copy16(anchors + idx * 4, &sA[idx * 4]);
  wait_async_all();
  __syncthreads();

  const int lane = tid & 31;

  // Pull each lane's anchor columns into registers; build raw Gram matrix.
  v4f A[NA][CH];
  float g00 = 0, g01 = 0, g02 = 0, g11 = 0, g12 = 0, g22 = 0;
#pragma unroll
  for (int c = 0; c < CH; ++c) {
    const int e = (c * 32 + lane) * 4;
    v4f a0 = *(const v4f*)&sA[0 * D + e];
    v4f a1 = *(const v4f*)&sA[1 * D + e];
    v4f a2 = *(const v4f*)&sA[2 * D + e];
    A[0][c] = a0; A[1][c] = a1; A[2][c] = a2;
    g00 += dot4(a0, a0); g01 += dot4(a0, a1); g02 += dot4(a0, a2);
    g11 += dot4(a1, a1); g12 += dot4(a1, a2); g22 += dot4(a2, a2);
  }
  g00 = wave_sum(g00); g01 = wave_sum(g01); g02 = wave_sum(g02);
  g11 = wave_sum(g11); g12 = wave_sum(g12); g22 = wave_sum(g22);

  const float i0 = 1.0f / fmaxf(sqrtf(g00), EPS_NORM);
  const float i1 = 1.0f / fmaxf(sqrtf(g11), EPS_NORM);
  const float i2 = 1.0f / fmaxf(sqrtf(g22), EPS_NORM);
  // Gram of the *normalized* anchors
  const float G00 = g00 * i0 * i0, G01 = g01 * i0 * i1, G02 = g02 * i0 * i2;
  const float G11 = g11 * i1 * i1, G12 = g12 * i1 * i2, G22 = g22 * i2 * i2;
#pragma unroll
  for (int c = 0; c < CH; ++c) {
    A[0][c] *= i0;
    A[1][c] *= i1;
    A[2][c] *= i2;
  }

  const int wave = (int)(blockIdx.x * (blockDim.x >> 5)) + (tid >> 5);
  const int nw   = (int)(gridDim.x * (blockDim.x >> 5));

  for (int row = wave; row < nrows; row += nw) {
    const float* hp = h0 + (size_t)row * D;

    // Prefetch next row of this wave's stream.
    if (row + nw < nrows) {
      const float* pp = h0 + (size_t)(row + nw) * D;
#pragma unroll
      for (int c = 0; c < CH; ++c)
        __builtin_prefetch(pp + (c * 32 + lane) * 4, 0, 1);
    }

    // Load row (coalesced, non-temporal: single-use stream), initial dots.
    v4f h[CH];
    float q = 0, p0 = 0, p1 = 0, p2 = 0;
#pragma unroll
    for (int c = 0; c < CH; ++c) {
      v4f v = __builtin_nontemporal_load((const v4f*)(hp + (c * 32 + lane) * 4));
      h[c] = v;
      q  += dot4(v, v);
      p0 += dot4(v, A[0][c]);
      p1 += dot4(v, A[1][c]);
      p2 += dot4(v, A[2][c]);
    }
    q  = wave_sum(q);
    p0 = wave_sum(p0); p1 = wave_sum(p1); p2 = wave_sum(p2);

    // h_t = a*h0 + b0*A0n + b1*A1n + b2*A2n — 6 steps collapse to scalars.
    float a = 1.0f, b0 = 0.0f, b1 = 0.0f, b2 = 0.0f;
    float d0 = p0, d1 = p1, d2 = p2, nsq = q;
#pragma unroll
    for (int s = 0; s < STEPS; ++s) {
      const float n   = sqrtf(nsq);
      const float inv = 1.0f / fmaxf(n, EPS_NORM);
      const float al0 = d0 * inv, al1 = d1 * inv, al2 = d2 * inv;
      const float mm  = fmaxf(al0, fmaxf(al1, al2));
      const float e0  = __expf(BETA * (al0 - mm));
      const float e1  = __expf(BETA * (al1 - mm));
      const float e2  = __expf(BETA * (al2 - mm));
      const float rs  = 1.0f / (e0 + e1 + e2);
      const float w0 = e0 * rs, w1 = e1 * rs, w2 = e2 * rs;
      const float sdot = w0 * al0 + w1 * al1 + w2 * al2;
      const float c0 = 1.0f - ALPHA * sdot * (n * inv);
      const float k0 = ALPHA * n * w0, k1 = ALPHA * n * w1, k2 = ALPHA * n * w2;
      // dots/norm propagate through the rank-3 update via the Gram matrix
      const float nd0 = c0 * d0 + k0 * G00 + k1 * G01 + k2 * G02;
      const float nd1 = c0 * d1 + k0 * G01 + k1 * G11 + k2 * G12;
      const float nd2 = c0 * d2 + k0 * G02 + k1 * G12 + k2 * G22;
      const float cross = k0 * d0 + k1 * d1 + k2 * d2;
      const float kk = k0 * (k0 * G00 + k1 * G01 + k2 * G02)
                     + k1 * (k0 * G01 + k1 * G11 + k2 * G12)
                     + k2 * (k0 * G02 + k1 * G12 + k2 * G22);
      const float nn = c0 * c0 * nsq + 2.0f * c0 * cross + kk;
      a = c0 * a; b0 = c0 * b0 + k0; b1 = c0 * b1 + k1; b2 = c0 * b2 + k2;
      // norm clip at MAX_NORM
      const float n2 = sqrtf(fmaxf(nn, 0.0f));
      const float t  = (n2 > MAX_NORM) ? (MAX_NORM / (n2 + EPS_CLIP)) : 1.0f;
      a *= t; b0 *= t; b1 *= t; b2 *= t;
      d0 = nd0 * t; d1 = nd1 * t; d2 = nd2 * t;
      nsq = nn * t * t;
    }

    // Materialize final h and store (coalesced, non-temporal: write-once).
    float* op = outh + (size_t)row * D;
#pragma unroll
    for (int c = 0; c < CH; ++c) {
      v4f v = h[c];
      v4f o = a * v + b0 * A[0][c] + b1 * A[1][c] + b2 * A[2][c];
      __builtin_nontemporal_store(o, (v4f*)(op + (c * 32 + lane) * 4));
    }
    // Final alignments come for free from the tracked dots.
    const float finv = 1.0f / fmaxf(sqrtf(nsq), EPS_NORM);
    if (lane < NA) {
      const float dd = (lane == 0) ? d0 : ((lane == 1) ? d1 : d2);
      outa[(size_t)row * NA + lane] = dd * finv;
    }
  }
}

extern "C" void kernel_launch(void* const* d_in, const int* in_sizes, int n_in,
                              void* d_out, int out_size, void* d_ws, size_t ws_size,
                              hipStream_t stream) {
  const float* h0      = (const float*)d_in[0];
  const float* anchors = (const float*)d_in[1];
  const int nrows = in_sizes[0] / D;           // 65536
  float* outh = (float*)d_out;
  float* outa = outh + (size_t)nrows * D;

  // one wave per row, ~4 rows per wave, 8 waves (256 threads) per block
  int waves  = (nrows + 3) / 4;
  int blocks = (waves + 7) / 8;
  if (blocks < 1) blocks = 1;
  livenium_kernel<<<blocks, 256, 0, stream>>>(h0, anchors, outh, outa, nrows);
}